// KNNAttentionLayer_81277961109951
// MI455X (gfx1250) — compile-verified
//
#include <hip/hip_runtime.h>

typedef __attribute__((ext_vector_type(2))) float v2f;
typedef __attribute__((ext_vector_type(8))) float v8f;

#define B_  4096
#define F_  8
#define V_  100000
#define D_  64
#define K_  16
#define NS  68           // padded LDS row stride (floats) -> rows hit distinct banks
#define WPB 8            // waves per block

__global__ __launch_bounds__(256) void knn_attn_wmma(const int* __restrict__ X,
                                                     const float* __restrict__ emb,
                                                     const int* __restrict__ adj,
                                                     float* __restrict__ out) {
    __shared__ float lds[WPB][K_ * NS + D_];   // per-wave: N tile (16 x 68) + self e (64)

    const int tid  = threadIdx.x;
    const int w    = tid >> 5;        // wave in block
    const int lane = tid & 31;
    const int h    = lane >> 4;       // half of wave (0 = lanes 0-15, 1 = lanes 16-31)
    const int col  = lane & 15;

    const int item = blockIdx.x * WPB + w;     // item in [0, B*F)
    const int f    = item & (F_ - 1);

    float* ldsN = lds[w];
    float* ldsE = lds[w] + K_ * NS;

    const size_t fV   = (size_t)f * V_;
    const int    x    = X[item];
    const float* srow = emb + (fV + (size_t)x) * D_;

    // ---- stage self embedding into LDS (64 floats; 2 per lane) ----
    *(float2*)(ldsE + lane * 2) = *(const float2*)(srow + lane * 2);

    // ---- gather 16 neighbor rows into LDS, 2 lanes per row, 128B contiguous per lane ----
    const int    nid  = adj[(fV + (size_t)x) * K_ + col];
    const float* nrow = emb + (fV + (size_t)nid) * D_ + h * 32;
    float*       drow = ldsN + col * NS + h * 32;
#pragma unroll
    for (int j = 0; j < 8; ++j) {
        float4 v = ((const float4*)nrow)[j];
        *(float4*)(drow + j * 4) = v;
    }

    // ---- scores: S[m] = sum_d N[m][d] * e[d]  via 16 chained WMMA f32 16x16x4 ----
    // A: lane (m = col): v0 = N[m][d0+2h], v1 = N[m][d0+2h+1]
    // B: broadcast e chunk across all 16 columns
    v8f cs = {};
#pragma unroll
    for (int d0 = 0; d0 < D_; d0 += 4) {
        v2f a = *(const v2f*)(ldsN + col * NS + d0 + 2 * h);
        v2f b = *(const v2f*)(ldsE + d0 + 2 * h);
        cs = __builtin_amdgcn_wmma_f32_16x16x4_f32(false, a, false, b,
                                                   (short)0, cs, false, false);
    }

    // ---- collect all 16 scores per lane (other half via shfl_xor 16) ----
    float sc[K_];
#pragma unroll
    for (int j = 0; j < 8; ++j) {
        float mine = cs[j];
        float oth  = __shfl_xor(mine, 16, 32);
        sc[j]     = h ? oth  : mine;   // rows 0-7
        sc[j + 8] = h ? mine : oth;    // rows 8-15
    }

    // ---- softmax over K=16 (every lane redundantly; keeps EXEC full for WMMA) ----
    float m = sc[0];
#pragma unroll
    for (int j = 1; j < K_; ++j) m = fmaxf(m, sc[j]);
    float wgt[K_];
    float sum = 0.f;
#pragma unroll
    for (int j = 0; j < K_; ++j) { wgt[j] = __expf(sc[j] - m); sum += wgt[j]; }
    const float inv = 1.0f / sum;
#pragma unroll
    for (int j = 0; j < K_; ++j) wgt[j] *= inv;

    // ---- weighted sum: out[n] = sum_k w[k] * N[k][n]  (4 accumulators x 4 K-steps) ----
    // A: w replicated across rows; B: N column chunk (16 cols), rows 4c4+2h, 4c4+2h+1
    v8f acc[4] = {};
#pragma unroll
    for (int g = 0; g < 4; ++g) {
#pragma unroll
        for (int c4 = 0; c4 < 4; ++c4) {
            v2f a;
            a.x = h ? wgt[4 * c4 + 2] : wgt[4 * c4 + 0];
            a.y = h ? wgt[4 * c4 + 3] : wgt[4 * c4 + 1];
            const float* bp = ldsN + (4 * c4 + 2 * h) * NS + g * 16 + col;
            v2f b;
            b.x = bp[0];
            b.y = bp[NS];
            acc[g] = __builtin_amdgcn_wmma_f32_16x16x4_f32(false, a, false, b,
                                                           (short)0, acc[g], false, false);
        }
    }

    // ---- residual + store: lanes 0-15 write chunks 0,1; lanes 16-31 write chunks 2,3 ----
    const size_t obase = (size_t)item * D_;
    const int ga = 2 * h, gb = 2 * h + 1;
    float r0 = (h ? acc[2][0] : acc[0][0]) + ldsE[ga * 16 + col];
    float r1 = (h ? acc[3][0] : acc[1][0]) + ldsE[gb * 16 + col];
    out[obase + ga * 16 + col] = r0;
    out[obase + gb * 16 + col] = r1;
}

extern "C" void kernel_launch(void* const* d_in, const int* in_sizes, int n_in,
                              void* d_out, int out_size, void* d_ws, size_t ws_size,
                              hipStream_t stream) {
    const int*   X   = (const int*)d_in[0];
    const float* emb = (const float*)d_in[1];
    const int*   adj = (const int*)d_in[2];
    float*       out = (float*)d_out;

    dim3 grid((B_ * F_) / WPB);   // 4096 blocks
    dim3 block(WPB * 32);         // 256 threads = 8 waves
    hipLaunchKernelGGL(knn_attn_wmma, grid, block, 0, stream, X, emb, adj, out);
}